// LLMLabelOnehot_67619965108953
// MI455X (gfx1250) — compile-verified
//
#include <hip/hip_runtime.h>
#include <hip/hip_bf16.h>

// LLMLabelOnehot: out[b,t,:] = 0; then for l in 0..L-1: out[b, t, label[b,l,t]] = probs[l]
// B=128, T=18, L=3, C=50000  -> out has 115.2M f32 (460.8 MB). Pure HBM-store bound.
//
// Kernel 1: non-temporal b128 zero-fill (stream > 2x L2, so NT avoids cache thrash).
// Kernel 2: one thread per (b,t); sequential l-loop inside the thread preserves
//           the reference's "later list overwrites earlier" semantics deterministically.

#define BT_B 128
#define BT_T 18
#define BT_L 3
#define BT_C 50000

typedef float v4f __attribute__((ext_vector_type(4)));  // native vector: OK for nontemporal builtin

__global__ void onehot_zero_fill_b128(v4f* __restrict__ out4, long long n4,
                                      float* __restrict__ out_tail, long long n_tail) {
    long long i = (long long)blockIdx.x * blockDim.x + threadIdx.x;
    long long stride = (long long)gridDim.x * blockDim.x;
    v4f z = {0.0f, 0.0f, 0.0f, 0.0f};
    for (; i < n4; i += stride) {
        // global_store_b128 with TH=NT on gfx1250: streaming store, bypass temporal caching
        __builtin_nontemporal_store(z, &out4[i]);
    }
    // generic tail (empty for 115,200,000 % 4 == 0, kept for safety)
    long long t = (long long)blockIdx.x * blockDim.x + threadIdx.x;
    if (t < n_tail) {
        __builtin_nontemporal_store(0.0f, &out_tail[t]);
    }
}

__global__ void onehot_scatter(const int* __restrict__ labels,   // [B, L, T] int
                               const float* __restrict__ probs,  // [L]
                               float* __restrict__ out) {        // [B, T, C]
    int idx = blockIdx.x * blockDim.x + threadIdx.x;             // one thread per (b,t)
    if (idx >= BT_B * BT_T) return;
    int b = idx / BT_T;
    int t = idx - b * BT_T;

    float p0 = probs[0];
    float p1 = probs[1];
    float p2 = probs[2];

    long long base = ((long long)b * BT_T + t) * (long long)BT_C;
    const int* lab = labels + ((long long)b * BT_L) * BT_T + t;  // stride T between lists

    // Sequential writes in one lane => deterministic "last list wins" on collisions.
    int c0 = lab[0 * BT_T];
    out[base + c0] = p0;
    int c1 = lab[1 * BT_T];
    out[base + c1] = p1;
    int c2 = lab[2 * BT_T];
    out[base + c2] = p2;
}

extern "C" void kernel_launch(void* const* d_in, const int* in_sizes, int n_in,
                              void* d_out, int out_size, void* d_ws, size_t ws_size,
                              hipStream_t stream) {
    (void)n_in; (void)d_ws; (void)ws_size; (void)in_sizes;

    const int*   labels = (const int*)d_in[0];   // LLM_label [B, L, T]
    const float* probs  = (const float*)d_in[1]; // [L]
    float*       out    = (float*)d_out;         // [B, T, C]

    const long long n  = (long long)out_size;    // 115,200,000
    const long long n4 = n >> 2;                 // 28,800,000 float4
    const long long n_tail = n - (n4 << 2);      // 0 here
    float* tail_ptr = out + (n4 << 2);

    // Fill: 256 threads/block, ~4 grid-stride iterations per thread.
    const int threads = 256;
    long long want_blocks = (n4 + (long long)threads * 4 - 1) / ((long long)threads * 4);
    int blocks = (int)(want_blocks > 0 ? want_blocks : 1);
    onehot_zero_fill_b128<<<blocks, threads, 0, stream>>>((v4f*)out, n4, tail_ptr, n_tail);

    // Scatter: B*T = 2304 threads.
    const int bt = BT_B * BT_T;
    onehot_scatter<<<(bt + 255) / 256, 256, 0, stream>>>(labels, probs, out);
}